// FactorizedJumpOperator_51281909514364
// MI455X (gfx1250) — compile-verified
//
#include <hip/hip_runtime.h>

#define LATENT_D   128
#define NUM_CHARTS 64
#define MAX_NORM   0.99f
#define MOB_EPS    1e-15f

typedef float v2f __attribute__((ext_vector_type(2)));
typedef float v8f __attribute__((ext_vector_type(8)));

__device__ __forceinline__ float waveSum(float v) {
#pragma unroll
    for (int o = 16; o > 0; o >>= 1) v += __shfl_xor(v, o, 32);
    return v;
}

// ---------------------------------------------------------------------------
// k0: zero the two 64-entry histograms (contiguous: cntT[64] then cntS[64])
// ---------------------------------------------------------------------------
__global__ void jump_zero(int* __restrict__ p) {
    p[threadIdx.x] = 0;
}

// ---------------------------------------------------------------------------
// k1: z = proj(z_n); zg = mobius_add(-proj(centers[src]), z); histograms.
// One wave per row, 4 elements per lane (wave32), shuffle reductions.
// ---------------------------------------------------------------------------
__global__ __launch_bounds__(256)
void jump_pre(const float* __restrict__ z_n,
              const float* __restrict__ centers,
              const int*   __restrict__ src_idx,
              const int*   __restrict__ tgt_idx,
              float*       __restrict__ zg,
              int* __restrict__ cntT, int* __restrict__ cntS, int B)
{
    const int wave = threadIdx.x >> 5, lane = threadIdx.x & 31;
    const int r = blockIdx.x * 8 + wave;
    if (r >= B) return;

    float y[4], x[4];
    float s = 0.f;
#pragma unroll
    for (int j = 0; j < 4; ++j) {
        y[j] = z_n[(size_t)r * LATENT_D + lane + 32 * j];
        s += y[j] * y[j];
    }
    s = waveSum(s);
    const float nz = sqrtf(s);
    const float sz = (nz > MAX_NORM) ? MAX_NORM / nz : 1.f;
    const float y2 = s * sz * sz;
#pragma unroll
    for (int j = 0; j < 4; ++j) y[j] *= sz;

    const int si = src_idx[r];
    float cs = 0.f;
#pragma unroll
    for (int j = 0; j < 4; ++j) {
        x[j] = centers[(size_t)si * LATENT_D + lane + 32 * j];
        cs += x[j] * x[j];
    }
    cs = waveSum(cs);
    const float ncs = sqrtf(cs);
    const float sc  = (ncs > MAX_NORM) ? MAX_NORM / ncs : 1.f;
    const float x2  = cs * sc * sc;
#pragma unroll
    for (int j = 0; j < 4; ++j) x[j] = -x[j] * sc;   // mobius_add(-c_source, z)

    float xy = 0.f;
#pragma unroll
    for (int j = 0; j < 4; ++j) xy += x[j] * y[j];
    xy = waveSum(xy);

    const float ka  = 1.f + 2.f * xy + y2;   // (1 + 2c xy + c y2), c = 1
    const float kb  = 1.f - x2;              // (1 - c x2)
    const float den = fmaxf(1.f + 2.f * xy + x2 * y2, MOB_EPS);
    const float inv = 1.f / den;
#pragma unroll
    for (int j = 0; j < 4; ++j)
        zg[(size_t)r * LATENT_D + lane + 32 * j] = (ka * x[j] + kb * y[j]) * inv;

    if (lane == 0) {
        atomicAdd(&cntT[tgt_idx[r]], 1);
        atomicAdd(&cntS[si], 1);
    }
}

// ---------------------------------------------------------------------------
// k2: exclusive scan of both 64-entry histograms -> starts and cursors
// ---------------------------------------------------------------------------
__global__ void jump_scan(const int* __restrict__ cntT, const int* __restrict__ cntS,
                          int* __restrict__ startT, int* __restrict__ startS,
                          int* __restrict__ curT,   int* __restrict__ curS)
{
    if (threadIdx.x != 0) return;
    int a = 0, b = 0;
    for (int i = 0; i < NUM_CHARTS; ++i) {
        startT[i] = a; curT[i] = a; a += cntT[i];
        startS[i] = b; curS[i] = b; b += cntS[i];
    }
}

// ---------------------------------------------------------------------------
// k3: scatter row indices into per-chart segments (counting-sort finish)
// ---------------------------------------------------------------------------
__global__ void jump_scatter(const int* __restrict__ src_idx, const int* __restrict__ tgt_idx,
                             int* __restrict__ curT, int* __restrict__ curS,
                             int* __restrict__ permT, int* __restrict__ permS, int B)
{
    const int b = blockIdx.x * blockDim.x + threadIdx.x;
    if (b >= B) return;
    const int p = atomicAdd(&curT[tgt_idx[b]], 1); permT[p] = b;
    const int q = atomicAdd(&curS[src_idx[b]], 1); permS[q] = b;
}

// ---------------------------------------------------------------------------
// k4/k5: per-chart grouped GEMM with V_WMMA_F32_16X16X4_F32.
//   TRANSPOSE=1:  Y[r,:] = R_chart   @ X[r,:]   (B-fragment from R^T)
//   TRANSPOSE=0:  Y[r,:] = R_chart^T @ X[r,:]   (B-fragment from R, coalesced)
// Block = 256 threads = 8 waves; wave w owns output columns [16w, 16w+16).
// A tile (16 gathered rows x 128) staged in LDS, pitch 132 -> conflict-free
// fragment reads (bank = 4*m + k distinct across all 32 lanes).
// TRANSPOSE is a template parameter so each instantiation is straight-line:
// the inner loop is exactly {global_load_b64, ds_load_b64, v_wmma} per K-step.
// R is L2-resident (4 MB total, 192 MB L2) -> no prefetch needed.
// ---------------------------------------------------------------------------
#define A_PITCH 132

template <int TRANSPOSE>
__global__ __launch_bounds__(256)
void jump_gemm(const float* __restrict__ X,
               float*       __restrict__ Y,
               const float* __restrict__ Rall,
               const int*   __restrict__ cnt,
               const int*   __restrict__ start,
               const int*   __restrict__ perm)
{
    __shared__ float Als[16 * A_PITCH];
    __shared__ int   rowIdx[16];

    const int chart = blockIdx.x;
    const int tile  = blockIdx.y;
    const int count = cnt[chart];
    if (tile * 16 >= count) return;                   // block-uniform exit
    const int seg  = start[chart] + tile * 16;
    const int rows = min(16, count - tile * 16);

    const int t = threadIdx.x;
    if (t < 16) rowIdx[t] = (t < rows) ? perm[seg + t] : 0;
    __syncthreads();

    // Gather A tile (zero-fill padding rows)
#pragma unroll
    for (int e = 0; e < 8; ++e) {
        const int idx = e * 256 + t;
        const int r = idx >> 7, col = idx & 127;
        float val = 0.f;
        if (r < rows) val = X[(size_t)rowIdx[r] * LATENT_D + col];
        Als[r * A_PITCH + col] = val;
    }
    __syncthreads();

    const float* Rc = Rall + (size_t)chart * LATENT_D * LATENT_D;
    const int lane  = t & 31;
    const int wave  = t >> 5;
    const int nbase = wave * 16;
    const int m     = lane & 15;
    const int kh    = (lane >> 4) << 1;               // 0 (lanes 0-15) / 2 (16-31)

    // Base pointers for B-fragment loads (constant per lane)
    const float* Brow = TRANSPOSE ? (Rc + (size_t)(nbase + m) * LATENT_D + kh)  // walk along K
                                  : (Rc + (size_t)kh * LATENT_D + nbase + m);  // walk down rows
    const float* Aptr = Als + m * A_PITCH + kh;

    v8f acc = {0.f, 0.f, 0.f, 0.f, 0.f, 0.f, 0.f, 0.f};
#pragma unroll 8
    for (int kk = 0; kk < LATENT_D; kk += 4) {
        v2f a, b;
        a.x = Aptr[kk];
        a.y = Aptr[kk + 1];
        if (TRANSPOSE) {
            // b = {R[n][kk+kh], R[n][kk+kh+1]} : contiguous 8B per lane
            b.x = Brow[kk];
            b.y = Brow[kk + 1];
        } else {
            // b = {R[kk+kh][n], R[kk+kh+1][n]} : coalesced across lanes
            b.x = Brow[(size_t)kk * LATENT_D];
            b.y = Brow[(size_t)(kk + 1) * LATENT_D];
        }
        // (neg_a, A, neg_b, B, c_mod, C, reuse_a, reuse_b)
        acc = __builtin_amdgcn_wmma_f32_16x16x4_f32(false, a, false, b,
                                                    (short)0, acc, false, false);
    }

    // C/D layout: VGPR v -> (M = v [+8 for upper half], N = lane&15)
    const int nc   = nbase + m;
    const int mofs = (lane >> 4) << 3;
#pragma unroll
    for (int v = 0; v < 8; ++v) {
        const int M = v + mofs;
        if (M < rows) Y[(size_t)rowIdx[M] * LATENT_D + nc] = acc[v];
    }
}

// ---------------------------------------------------------------------------
// k6: out = proj( mobius_add( proj(centers[tgt]), v ) )
// ---------------------------------------------------------------------------
__global__ __launch_bounds__(256)
void jump_post(const float* __restrict__ vrow,
               const float* __restrict__ centers,
               const int*   __restrict__ tgt_idx,
               float*       __restrict__ out, int B)
{
    const int wave = threadIdx.x >> 5, lane = threadIdx.x & 31;
    const int r = blockIdx.x * 8 + wave;
    if (r >= B) return;

    float x[4], y[4];
    const int ti = tgt_idx[r];
    float cs = 0.f;
#pragma unroll
    for (int j = 0; j < 4; ++j) {
        x[j] = centers[(size_t)ti * LATENT_D + lane + 32 * j];
        cs += x[j] * x[j];
    }
    cs = waveSum(cs);
    const float ncs = sqrtf(cs);
    const float sc  = (ncs > MAX_NORM) ? MAX_NORM / ncs : 1.f;
    const float x2  = cs * sc * sc;
#pragma unroll
    for (int j = 0; j < 4; ++j) x[j] *= sc;           // +c_target this time

    float y2 = 0.f;
#pragma unroll
    for (int j = 0; j < 4; ++j) {
        y[j] = vrow[(size_t)r * LATENT_D + lane + 32 * j];
        y2 += y[j] * y[j];
    }
    y2 = waveSum(y2);

    float xy = 0.f;
#pragma unroll
    for (int j = 0; j < 4; ++j) xy += x[j] * y[j];
    xy = waveSum(xy);

    const float ka  = 1.f + 2.f * xy + y2;
    const float kb  = 1.f - x2;
    const float den = fmaxf(1.f + 2.f * xy + x2 * y2, MOB_EPS);
    const float inv = 1.f / den;

    float o[4], so = 0.f;
#pragma unroll
    for (int j = 0; j < 4; ++j) { o[j] = (ka * x[j] + kb * y[j]) * inv; so += o[j] * o[j]; }
    so = waveSum(so);
    const float no = sqrtf(so);
    const float sf = (no > MAX_NORM) ? MAX_NORM / no : 1.f;
#pragma unroll
    for (int j = 0; j < 4; ++j)
        out[(size_t)r * LATENT_D + lane + 32 * j] = o[j] * sf;
}

// ---------------------------------------------------------------------------
extern "C" void kernel_launch(void* const* d_in, const int* in_sizes, int n_in,
                              void* d_out, int out_size, void* d_ws, size_t ws_size,
                              hipStream_t stream)
{
    const float* z_n     = (const float*)d_in[0];
    const float* centers = (const float*)d_in[1];
    const float* rot     = (const float*)d_in[2];
    const int*   src     = (const int*)d_in[3];
    const int*   tgt     = (const int*)d_in[4];
    float*       out     = (float*)d_out;

    const int B = in_sizes[0] / LATENT_D;

    // Workspace layout: zg[B*128] | w[B*128] | int area
    float* zg = (float*)d_ws;
    float* w  = zg + (size_t)B * LATENT_D;
    int* ints = (int*)(w + (size_t)B * LATENT_D);
    int* cntT   = ints;          // 64
    int* cntS   = ints + 64;     // 64
    int* startT = ints + 128;    // 64
    int* startS = ints + 192;    // 64
    int* curT   = ints + 256;    // 64
    int* curS   = ints + 320;    // 64
    int* permT  = ints + 384;    // B
    int* permS  = permT + B;     // B

    jump_zero<<<1, 128, 0, stream>>>(cntT);  // cntT+cntS contiguous

    jump_pre<<<dim3((B + 7) / 8), 256, 0, stream>>>(
        z_n, centers, src, tgt, zg, cntT, cntS, B);

    jump_scan<<<1, 32, 0, stream>>>(cntT, cntS, startT, startS, curT, curS);

    jump_scatter<<<dim3((B + 255) / 256), 256, 0, stream>>>(
        src, tgt, curT, curS, permT, permS, B);

    dim3 ggrid(NUM_CHARTS, (B + 15) / 16);
    // Stage 1: w[r] = R_target @ zg[r]      (group by target)
    jump_gemm<1><<<ggrid, 256, 0, stream>>>(zg, w, rot, cntT, startT, permT);
    // Stage 2: zg[r] = R_source^T @ w[r]    (group by source; zg reused)
    jump_gemm<0><<<ggrid, 256, 0, stream>>>(w, zg, rot, cntS, startS, permS);

    jump_post<<<dim3((B + 7) / 8), 256, 0, stream>>>(zg, centers, tgt, out, B);
}